// DIST_loss_49331994362453
// MI455X (gfx1250) — compile-verified
//
#include <hip/hip_runtime.h>

typedef __attribute__((ext_vector_type(2))) float v2f;
typedef __attribute__((ext_vector_type(4))) float v4f;
typedef __attribute__((ext_vector_type(8))) float v8f;

// Exact 32-lane f32 sum using one V_WMMA_F32_16X16X4_F32.
// A-matrix (16x4): lane m (m<16) supplies A(m,0)=s via a.x, lanes 16..31 supply
// A(m,2)=s ; a.y -> K=1/3 slots = 0. B = ones(4x16).
// D(m,n) = s_m + s_{m+16} replicated over n; lane l's 8 C/D VGPRs hold row sums
// m = 8*(l/16)..8*(l/16)+7, so 7 VALU adds + one xor-16 shuffle finish the sum.
// Fixed add order -> deterministic. EXEC must be all-ones at every call site.
__device__ __forceinline__ float wave_sum32(float s) {
    v2f a; a.x = s;    a.y = 0.0f;
    v2f b; b.x = 1.0f; b.y = 1.0f;
    v8f c = {0.f, 0.f, 0.f, 0.f, 0.f, 0.f, 0.f, 0.f};
    c = __builtin_amdgcn_wmma_f32_16x16x4_f32(
        /*neg_a=*/false, a, /*neg_b=*/false, b,
        /*c_mod=*/(short)0, c, /*reuse_a=*/false, /*reuse_b=*/false);
    float r = ((c[0] + c[1]) + (c[2] + c[3])) + ((c[4] + c[5]) + (c[6] + c[7]));
    r += __shfl_xor(r, 16, 32);   // combine rows 0..7 half with rows 8..15 half
    return r;
}

// Block (256 threads = 8 wave32) reduction: WMMA wave sums -> LDS -> WMMA again.
__device__ __forceinline__ float block_sum256(float acc, float* smem8) {
    const int lane = threadIdx.x & 31;
    const int wid  = threadIdx.x >> 5;
    float ws = wave_sum32(acc);                 // all 8 waves, full EXEC
    if (lane == 0) smem8[wid] = ws;
    __syncthreads();
    float total = 0.0f;
    if (wid == 0) {
        float v = (lane < 8) ? smem8[lane] : 0.0f;
        total = wave_sum32(v);                  // wave 0 only, full EXEC within wave
    }
    return total;                               // valid in wave 0
}

__global__ void __launch_bounds__(256) dist_partial_kernel(
        const float* __restrict__ preds, const float* __restrict__ targets,
        float* __restrict__ partials, long long nf4, long long nfloat) {
    float acc = 0.0f;
    const long long stride = (long long)gridDim.x * blockDim.x;
    for (long long i = (long long)blockIdx.x * blockDim.x + threadIdx.x;
         i < nf4; i += stride) {
        // streaming b128 loads, non-temporal (256 MiB read-once > 192 MB L2)
        v4f p = __builtin_nontemporal_load((const v4f*)preds + i);
        v4f t = __builtin_nontemporal_load((const v4f*)targets + i);
        float dx0 = p.x - t.x, dy0 = p.y - t.y;
        float dx1 = p.z - t.z, dy1 = p.w - t.w;
        acc += __builtin_sqrtf(dx0 * dx0 + dy0 * dy0);
        acc += __builtin_sqrtf(dx1 * dx1 + dy1 * dy1);
    }
    // Generic tail (floats beyond nf4*4); empty for the reference shape.
    if (blockIdx.x == 0 && threadIdx.x == 0) {
        for (long long j = nf4 * 4; j + 1 < nfloat; j += 2) {
            float dx = preds[j]     - targets[j];
            float dy = preds[j + 1] - targets[j + 1];
            acc += __builtin_sqrtf(dx * dx + dy * dy);
        }
    }
    __shared__ float smem[8];
    float bs = block_sum256(acc, smem);
    if (threadIdx.x == 0) partials[blockIdx.x] = bs;
}

__global__ void __launch_bounds__(256) dist_final_kernel(
        const float* __restrict__ partials, int nPartials,
        float* __restrict__ out, float denom) {
    float acc = 0.0f;
    for (int i = threadIdx.x; i < nPartials; i += 256) acc += partials[i];
    __shared__ float smem[8];
    float bs = block_sum256(acc, smem);
    if (threadIdx.x == 0) out[0] = bs / denom;
}

extern "C" void kernel_launch(void* const* d_in, const int* in_sizes, int n_in,
                              void* d_out, int out_size, void* d_ws, size_t ws_size,
                              hipStream_t stream) {
    const float* preds   = (const float*)d_in[0];
    const float* targets = (const float*)d_in[1];
    float* outf = (float*)d_out;
    float* partials = (float*)d_ws;

    const long long nfloat  = (long long)in_sizes[0];   // 2*N
    const long long npoints = nfloat / 2;               // N
    const long long nf4     = nfloat / 4;               // float4 elements

    int blocks = 4096;                                   // 16 KiB of partials
    if ((size_t)blocks * sizeof(float) > ws_size)
        blocks = (int)(ws_size / sizeof(float));
    long long needed = (nf4 + 255) / 256;
    if (needed < 1) needed = 1;
    if ((long long)blocks > needed) blocks = (int)needed;
    if (blocks < 1) blocks = 1;

    dist_partial_kernel<<<blocks, 256, 0, stream>>>(preds, targets, partials,
                                                    nf4, nfloat);
    const float denom = (float)(npoints + 1);            // torch-faithful mean over N+1
    dist_final_kernel<<<1, 256, 0, stream>>>(partials, blocks, outf, denom);
}